// SAGE_58282706206971
// MI455X (gfx1250) — compile-verified
//
#include <hip/hip_runtime.h>
#include <math.h>

#define N_NODES 100000
#define N_EDGES 1000000
#define IN_F    128
#define HID_F   128
#define OUT_F   16

typedef float v2f __attribute__((ext_vector_type(2)));
typedef float v8f __attribute__((ext_vector_type(8)));

// ---------------- degree / normalizer ----------------
__global__ void zero_f32(float* __restrict__ p, int n) {
    int i = blockIdx.x * blockDim.x + threadIdx.x;
    if (i < n) p[i] = 0.0f;
}

__global__ void deg_accum(const int* __restrict__ dst, float* __restrict__ deg, int nE) {
    int i = blockIdx.x * blockDim.x + threadIdx.x;
    if (i < nE) atomicAdd(&deg[dst[i]], 1.0f);
}

__global__ void make_inv(float* __restrict__ deg, int n) {
    int i = blockIdx.x * blockDim.x + threadIdx.x;
    if (i < n) deg[i] = 1.0f / (deg[i] + 1.0f);
}

// ---------------- aggregation ----------------
__global__ void copy_f4(float4* __restrict__ out, const float4* __restrict__ in, long n4) {
    long i = (long)blockIdx.x * blockDim.x + threadIdx.x;
    if (i < n4) out[i] = in[i];
}

// one wave per edge; lane l moves features [4l, 4l+3] (512B coalesced gather)
__global__ void edge_scatter(const float* __restrict__ H, float* __restrict__ AGG,
                             const int* __restrict__ src, const int* __restrict__ dst, int nE) {
    int gt = blockIdx.x * blockDim.x + threadIdx.x;
    int e = gt >> 5;
    if (e >= nE) return;
    int lane = threadIdx.x & 31;
    long s = src[e];
    long d = dst[e];
    const float4* hv = (const float4*)(H + s * HID_F);
    float4 v = hv[lane];
    float* o = AGG + d * HID_F + lane * 4;
    atomicAdd(o + 0, v.x);
    atomicAdd(o + 1, v.y);
    atomicAdd(o + 2, v.z);
    atomicAdd(o + 3, v.w);
}

// ---------------- WMMA GEMM: out = act(diag(inv) * A * W + b) ----------------
// A: (N_NODES x 128) row-major. W staged in LDS K-pair interleaved:
//   wlds[((k>>1)*NCOL + n)*2 + (k&1)] = W[k][n]
// so a B fragment {W[ka][n], W[ka+1][n]} is one contiguous ds_load_b64 into an
// even-aligned VGPR pair (no shuffle movs). One wave owns a 16-row M panel and
// computes all NCOL/16 column tiles, reusing each A fragment across N.
template <int NCOL, bool RELU>
__launch_bounds__(256)
__global__ void gemm_wmma_f32(const float* __restrict__ A, const float* __restrict__ inv,
                              const float* __restrict__ W, const float* __restrict__ bias,
                              float* __restrict__ out) {
    constexpr int TN = NCOL / 16;
    __shared__ float wlds[HID_F * NCOL];   // 64KB for NCOL=128, 8KB for NCOL=16

    // cooperative W stage with K-pair interleave (all threads, before any exit)
    for (int idx = threadIdx.x; idx < HID_F * NCOL; idx += blockDim.x) {
        const int r = idx / NCOL;          // k index
        const int c = idx % NCOL;          // n index
        wlds[(((r >> 1) * NCOL) + c) * 2 + (r & 1)] = W[idx];
    }
    __syncthreads();

    const int wid = blockIdx.x * (blockDim.x >> 5) + (threadIdx.x >> 5);
    if (wid >= N_NODES / 16) return;       // wave-uniform: EXEC all-ones at each WMMA

    const int lane = threadIdx.x & 31;
    const int half = lane >> 4;            // selects K pair {0,1} vs {2,3}
    const int l15  = lane & 15;
    const int kh   = half * 2;

    const int m = wid * 16 + l15;          // A-fragment row for this lane
    const float sc = inv[m];               // fused 1/(deg+1) row scaling
    const float* __restrict__ arow = A + (long)m * HID_F;
    const float2* __restrict__ wp = (const float2*)wlds;

    v8f c[TN];
    #pragma unroll
    for (int t = 0; t < TN; ++t) c[t] = (v8f){};

    // software-pipelined A fetch: raw pair for step k is loaded one step ahead
    float2 araw = *(const float2*)(arow + kh);

    #pragma unroll 4
    for (int k = 0; k < HID_F; k += 4) {
        v2f a;
        a.x = araw.x * sc;                 // A 16x4: VGPR0=K(ka), VGPR1=K(ka+1)
        a.y = araw.y * sc;
        if (k + 4 < HID_F)                 // prefetch next A pair before the WMMAs
            araw = *(const float2*)(arow + k + 4 + kh);

        const int kp = (k >> 1) + half;    // K-pair row in interleaved LDS image
        #pragma unroll
        for (int t = 0; t < TN; ++t) {
            const float2 bw = wp[kp * NCOL + t * 16 + l15];
            v2f b;
            b.x = bw.x;
            b.y = bw.y;
            c[t] = __builtin_amdgcn_wmma_f32_16x16x4_f32(
                       /*neg_a=*/false, a, /*neg_b=*/false, b,
                       /*c_mod=*/(short)0, c[t], /*reuse_a=*/false, /*reuse_b=*/false);
        }
    }

    #pragma unroll
    for (int t = 0; t < TN; ++t) {
        const int n = t * 16 + l15;
        const float bv = bias[n];
        #pragma unroll
        for (int r = 0; r < 8; ++r) {      // C/D: VGPR r holds rows r and r+8
            const int mo = wid * 16 + r + half * 8;
            float v = c[t][r] + bv;
            if (RELU) v = fmaxf(v, 0.0f);
            out[(long)mo * NCOL + n] = v;
        }
    }
}

// ---------------- log-softmax over 16 logits, in place ----------------
__global__ void logsoftmax16(float* __restrict__ z, int n) {
    int i = blockIdx.x * blockDim.x + threadIdx.x;
    if (i >= n) return;
    float* row = z + (long)i * OUT_F;
    float v[OUT_F];
    float mx = -3.402823466e+38f;
    #pragma unroll
    for (int j = 0; j < OUT_F; ++j) { v[j] = row[j]; mx = fmaxf(mx, v[j]); }
    float s = 0.0f;
    #pragma unroll
    for (int j = 0; j < OUT_F; ++j) s += expf(v[j] - mx);
    const float lse = mx + logf(s);
    #pragma unroll
    for (int j = 0; j < OUT_F; ++j) row[j] = v[j] - lse;
}

// ---------------- host launch ----------------
extern "C" void kernel_launch(void* const* d_in, const int* in_sizes, int n_in,
                              void* d_out, int out_size, void* d_ws, size_t ws_size,
                              hipStream_t stream) {
    const float* x   = (const float*)d_in[0];
    const int*   src = (const int*)d_in[1];   // JAX default config: int64 -> int32
    const int*   dst = (const int*)d_in[2];
    const float* Wl[5] = {(const float*)d_in[3], (const float*)d_in[5], (const float*)d_in[7],
                          (const float*)d_in[9], (const float*)d_in[11]};
    const float* bl[5] = {(const float*)d_in[4], (const float*)d_in[6], (const float*)d_in[8],
                          (const float*)d_in[10], (const float*)d_in[12]};

    float* AGG = (float*)d_ws;                         // 51.2 MB
    float* H   = AGG + (size_t)N_NODES * HID_F;        // 51.2 MB
    float* INV = H   + (size_t)N_NODES * HID_F;        // 0.4 MB

    // degree -> 1/(deg+1), computed fresh every call (deterministic)
    zero_f32 <<<(N_NODES + 255) / 256, 256, 0, stream>>>(INV, N_NODES);
    deg_accum<<<(N_EDGES + 255) / 256, 256, 0, stream>>>(dst, INV, N_EDGES);
    make_inv <<<(N_NODES + 255) / 256, 256, 0, stream>>>(INV, N_NODES);

    const long n4 = (long)N_NODES * HID_F / 4;
    const int copyBlocks = (int)((n4 + 255) / 256);
    const int scatBlocks = (int)(((long)N_EDGES * 32 + 255) / 256);
    const int mPanels    = N_NODES / 16;               // 6250, exact
    const int gemmBlocks = (mPanels + 7) / 8;          // 8 waves / 256-thread block

    const float* hcur = x;
    for (int l = 0; l < 4; ++l) {
        copy_f4     <<<copyBlocks, 256, 0, stream>>>((float4*)AGG, (const float4*)hcur, n4);
        edge_scatter<<<scatBlocks, 256, 0, stream>>>(hcur, AGG, src, dst, N_EDGES);
        gemm_wmma_f32<HID_F, true><<<gemmBlocks, 256, 0, stream>>>(AGG, INV, Wl[l], bl[l], H);
        hcur = H;
    }
    copy_f4     <<<copyBlocks, 256, 0, stream>>>((float4*)AGG, (const float4*)hcur, n4);
    edge_scatter<<<scatBlocks, 256, 0, stream>>>(hcur, AGG, src, dst, N_EDGES);
    gemm_wmma_f32<OUT_F, false><<<gemmBlocks, 256, 0, stream>>>(AGG, INV, Wl[4], bl[4], (float*)d_out);
    logsoftmax16<<<(N_NODES + 255) / 256, 256, 0, stream>>>((float*)d_out, N_NODES);
}